// Encoder_80255758893556
// MI455X (gfx1250) — compile-verified
//
#include <hip/hip_runtime.h>

// Native vector types so __builtin_nontemporal_load/store lowers to single
// b128 global instructions with TH=NT (HIP's float4 is a struct and would not).
typedef float v4f __attribute__((ext_vector_type(4)));
typedef int   v4i __attribute__((ext_vector_type(4)));

namespace {
// Match the JAX reference exactly: CPR and DT*CPR are computed in double
// precision on the host side of the reference, then demoted to f32 when they
// meet the f32 arrays (jax x64 disabled => f64 scalars cast to f32).
constexpr double kPi    = 3.14159265358979323846264338327950288;
constexpr double kCPRd  = 2048.0 / (2.0 * kPi);       // clicks per radian
constexpr float  kScale = (float)(0.1 * kCPRd);       // DT * CPR, f32-rounded
// Divisor (CPR * DT) is the same double product -> same f32 constant.
}

struct EncOut { float out; float rem; };

// One element of the reference, with rounding forced to match un-fused XLA
// elementwise ops (HIP defaults to -ffp-contract=fast, so we must use the
// *_rn intrinsics to forbid mul+add -> fma contraction, and a true IEEE
// divide rather than a reciprocal multiply).
__device__ __forceinline__ EncOut encoder1(float w, float r, float c, int s) {
    float temp = __fadd_rn(__fmul_rn(w, kScale), r);  // ws*(DT*CPR) + rem
    float nc   = truncf(temp);                        // jnp.trunc
    float no   = __fdiv_rn(nc, kScale);               // number_clicks/(CPR*DT)
    float nr   = __fsub_rn(temp, nc);

    bool nominal = (s == 0);
    bool off     = (s == 1);
    bool stuck   = (s == 2);

    EncOut e;
    e.out = nominal ? no : (stuck ? c : 0.0f);
    e.rem = nominal ? nr : (off ? 0.0f : r);
    return e;
}

// Vectorized main path: 4 elements per lane => b128 loads/stores, wave32
// moves full 512B per stream per instruction. Non-temporal hints because the
// 402MB stream exceeds the 192MB L2 and has zero reuse.
__global__ __launch_bounds__(256) void encoder_vec4(
    const v4f* __restrict__ ws, const v4f* __restrict__ rem,
    const v4f* __restrict__ conv, const v4i* __restrict__ st,
    v4f* __restrict__ out, v4f* __restrict__ outRem, int n4)
{
    int i = blockIdx.x * 256 + threadIdx.x;
    if (i >= n4) return;

    // gfx1250 global_prefetch_b8: pull ~256KB/stream ahead into the cache
    // hierarchy. One lane per 128B line (8 x 16B) to avoid redundant requests.
    constexpr int kPF = 16384;                 // float4 elements ahead
    if (((i & 7) == 0) && (i + kPF < n4)) {
        __builtin_prefetch((const void*)(ws   + i + kPF), 0, 0);
        __builtin_prefetch((const void*)(rem  + i + kPF), 0, 0);
        __builtin_prefetch((const void*)(conv + i + kPF), 0, 0);
        __builtin_prefetch((const void*)(st   + i + kPF), 0, 0);
    }

    v4f w = __builtin_nontemporal_load(ws   + i);
    v4f r = __builtin_nontemporal_load(rem  + i);
    v4f c = __builtin_nontemporal_load(conv + i);
    v4i s = __builtin_nontemporal_load(st   + i);

    v4f o, ro;
    EncOut e0 = encoder1(w.x, r.x, c.x, s.x); o.x = e0.out; ro.x = e0.rem;
    EncOut e1 = encoder1(w.y, r.y, c.y, s.y); o.y = e1.out; ro.y = e1.rem;
    EncOut e2 = encoder1(w.z, r.z, c.z, s.z); o.z = e2.out; ro.z = e2.rem;
    EncOut e3 = encoder1(w.w, r.w, c.w, s.w); o.w = e3.out; ro.w = e3.rem;

    __builtin_nontemporal_store(o,  out    + i);
    __builtin_nontemporal_store(ro, outRem + i);
}

// Scalar fallback for sizes not divisible by 4 (keeps kernel_launch general).
__global__ __launch_bounds__(256) void encoder_scalar(
    const float* __restrict__ ws, const float* __restrict__ rem,
    const float* __restrict__ conv, const int* __restrict__ st,
    float* __restrict__ out, float* __restrict__ outRem, int n)
{
    int i = blockIdx.x * 256 + threadIdx.x;
    if (i >= n) return;
    EncOut e = encoder1(ws[i], rem[i], conv[i], st[i]);
    out[i]    = e.out;
    outRem[i] = e.rem;
}

extern "C" void kernel_launch(void* const* d_in, const int* in_sizes, int n_in,
                              void* d_out, int out_size, void* d_ws, size_t ws_size,
                              hipStream_t stream) {
    const float* ws   = (const float*)d_in[0];   // wheel_speeds
    const float* rem  = (const float*)d_in[1];   // remaining_clicks
    const float* conv = (const float*)d_in[2];   // converted
    const int*   st   = (const int*)d_in[3];     // rw_signal_state

    int n = in_sizes[0];
    float* out    = (float*)d_out;       // new_output      [0, n)
    float* outRem = out + n;             // new_remaining   [n, 2n)

    if (n <= 0) return;

    if ((n & 3) == 0) {
        int n4 = n >> 2;
        int blocks = (n4 + 255) / 256;
        encoder_vec4<<<blocks, 256, 0, stream>>>(
            (const v4f*)ws, (const v4f*)rem, (const v4f*)conv, (const v4i*)st,
            (v4f*)out, (v4f*)outRem, n4);
    } else {
        int blocks = (n + 255) / 256;
        encoder_scalar<<<blocks, 256, 0, stream>>>(ws, rem, conv, st,
                                                   out, outRem, n);
    }
}